// SpaceTimeMetricModule_6390911336868
// MI455X (gfx1250) — compile-verified
//
#include <hip/hip_runtime.h>
#include <hip/hip_bf16.h>

// Schwarzschild metric: out[i] = diag(-(1-RS/r), 1/(1-RS/r), r^2, r^2 sin^2(theta))
// embedded in a 4x4 matrix (16 floats per row, 12 zeros).
//
// Pure streaming kernel: ~64 MB read + 256 MB write, HBM-bound (~13.7 us floor
// at 23.3 TB/s). One thread per output float4 chunk -> every global_store_b128
// is fully coalesced across the wave32. Output (256 MB) > L2 (192 MB) and is
// write-once, so stores carry the non-temporal hint (th:TH_STORE_NT).
// Divisions use v_rcp_f32 + one Newton step (3 VALU each instead of the ~12
// VALU IEEE div sequence) to keep VALU time well under the HBM floor.

typedef float v4f __attribute__((ext_vector_type(4)));

#define RS 2.0f

// 1/a via v_rcp_f32 + one Newton-Raphson refinement (~0.5-1 ulp).
__device__ __forceinline__ float fast_rcp(float a) {
    float x = __builtin_amdgcn_rcpf(a);
    float e = __builtin_fmaf(-a, x, 1.0f);   // 1 - a*x
    return __builtin_fmaf(x, e, x);          // x*(2 - a*x)
}

__global__ __launch_bounds__(256) void
schwarzschild_metric_kernel(const v4f* __restrict__ in,  // N rows: (t, r, theta, phi)
                            v4f* __restrict__ out,       // 4N chunks of 4 floats
                            int n_chunks)                // = 4 * N
{
    int t = blockIdx.x * blockDim.x + threadIdx.x;
    if (t >= n_chunks) return;

    int row = t >> 2;   // lanes 0..3 of each group of 4 share one input row
    int q   = t & 3;    // which 4-float chunk of the 16-float output row

    v4f rec = in[row];          // global_load_b128, groups of 4 lanes broadcast
    float r     = rec.y;
    float theta = rec.z;

    float rinv = fast_rcp(r);
    float f    = __builtin_fmaf(-RS, rinv, 1.0f);  // 1 - 2/r
    float gtt  = -f;
    float grr  = fast_rcp(f);
    float r2   = r * r;
    float s    = __sinf(theta);                    // v_mul(1/2pi) + v_sin_f32
    float gpp  = r2 * s * s;

    // Diagonal element 5q of the 16-float row lands in chunk q at position q.
    v4f o;
    o.x = (q == 0) ? gtt : 0.0f;
    o.y = (q == 1) ? grr : 0.0f;
    o.z = (q == 2) ? r2  : 0.0f;
    o.w = (q == 3) ? gpp : 0.0f;

    // Non-temporal streaming store: global_store_b128 ... th:TH_STORE_NT
    __builtin_nontemporal_store(o, out + t);
}

extern "C" void kernel_launch(void* const* d_in, const int* in_sizes, int n_in,
                              void* d_out, int out_size, void* d_ws, size_t ws_size,
                              hipStream_t stream) {
    const float* in = (const float*)d_in[0];
    float* out = (float*)d_out;

    int n_rows   = in_sizes[0] / 4;   // 4,000,000
    int n_chunks = n_rows * 4;        // 16,000,000 float4 chunks (== out_size/4)

    const int block = 256;            // 8 wave32 waves per block
    int grid = (n_chunks + block - 1) / block;

    schwarzschild_metric_kernel<<<grid, block, 0, stream>>>(
        (const v4f*)in, (v4f*)out, n_chunks);
}